// ItemGNNEncoder_48816598286983
// MI455X (gfx1250) — compile-verified
//
#include <hip/hip_runtime.h>

#define N_NODES 100000
#define NE      1600000
#define F       64

typedef __attribute__((ext_vector_type(2))) float v2f;
typedef __attribute__((ext_vector_type(8))) float v8f;

__device__ __forceinline__ v8f wmma_f32(v2f a, v2f b, v8f c) {
    // D = A(16x4 f32) * B(4x16 f32) + C(16x16 f32)
    return __builtin_amdgcn_wmma_f32_16x16x4_f32(
        /*neg_a=*/false, a, /*neg_b=*/false, b,
        /*c_mod=*/(short)0, c, /*reuse_a=*/false, /*reuse_b=*/false);
}

// Stage W (row-major 64x64) into LDS pre-swizzled into WMMA B-fragment pairs:
// sW2[(kk*2 + half)*64 + n] = { W[kk*4 + 2*half][n], W[kk*4 + 2*half + 1][n] }
// so each B fragment is ONE aligned ds_load_b64 into an even VGPR pair.
__device__ __forceinline__ void stage_w_pairs(v2f* __restrict__ sW2,
                                              const float* __restrict__ w) {
    for (int p = threadIdx.x; p < 32 * F; p += 256) {
        const int n  = p & 63;
        const int rp = p >> 6;                    // rp = kk*2 + half
        const int kb = (rp >> 1) * 4 + (rp & 1) * 2;
        v2f t;
        t.x = w[kb * F + n];
        t.y = w[(kb + 1) * F + n];
        sW2[p] = t;
    }
}

// ---------------- utility kernels ----------------

__global__ void zero_kernel(float* __restrict__ p, long n) {
    long i = (long)blockIdx.x * blockDim.x + threadIdx.x;
    if (i < n) p[i] = 0.0f;
}

__global__ void count_kernel(const long long* __restrict__ dst, float* __restrict__ cnt) {
    int e = blockIdx.x * blockDim.x + threadIdx.x;
    if (e < NE) atomicAdd(cnt + (long)dst[e], 1.0f);
}

__global__ void inv_kernel(const float* __restrict__ cnt, float* __restrict__ inv) {
    int i = blockIdx.x * blockDim.x + threadIdx.x;
    if (i < N_NODES) inv[i] = 1.0f / fmaxf(cnt[i], 1.0f);
}

// Edge scatter: 16 threads per edge, each moves a float4 of the 64-wide row.
// x and summed are L2-resident (25.6 MB each vs 192 MB L2) -> atomics resolve in L2.
__global__ void scatter_kernel(const float* __restrict__ x,
                               const long long* __restrict__ src,
                               const long long* __restrict__ dst,
                               float* __restrict__ summed) {
    long t = (long)blockIdx.x * blockDim.x + threadIdx.x;
    if (t >= (long)NE * 16) return;
    int e = (int)(t >> 4);
    int c = (int)(t & 15);
    // stream-ahead prefetch of the edge list (global_prefetch_b8)
    if ((t & 15) == 0 && e + 8192 < NE) {
        __builtin_prefetch(src + e + 8192, 0, 0);
        __builtin_prefetch(dst + e + 8192, 0, 0);
    }
    long s = (long)src[e];
    long d = (long)dst[e];
    const float4 v = *(const float4*)(x + s * F + c * 4);
    float* o = summed + d * F + c * 4;
    atomicAdd(o + 0, v.x);
    atomicAdd(o + 1, v.y);
    atomicAdd(o + 2, v.z);
    atomicAdd(o + 3, v.w);
}

// ---------------- fused SAGE conv GEMM ----------------
// out = relu( (summed*inv) @ wl + bias + xin @ wr )
// Block = 256 threads = 8 waves; each wave owns a 16-row x 64-col output tile.
__global__ void __launch_bounds__(256)
sage_gemm_kernel(const float* __restrict__ summed, const float* __restrict__ inv,
                 const float* __restrict__ xin,
                 const float* __restrict__ wl, const float* __restrict__ wr,
                 const float* __restrict__ bias, float* __restrict__ out) {
    __shared__ v2f sWL2[32 * F];
    __shared__ v2f sWR2[32 * F];
    stage_w_pairs(sWL2, wl);
    stage_w_pairs(sWR2, wr);
    __syncthreads();

    const int lane = threadIdx.x & 31;
    const int wave = threadIdx.x >> 5;
    const int half = lane >> 4;   // 0: lanes 0-15, 1: lanes 16-31
    const int lr   = lane & 15;
    const int row0 = blockIdx.x * 128 + wave * 16;

    int rowA = row0 + lr;
    if (rowA > N_NODES - 1) rowA = N_NODES - 1;   // clamp loads, keep EXEC full
    const float  sc = inv[rowA];
    const float* aL = summed + (size_t)rowA * F + 2 * half;  // K = k0 + 2*half + {0,1}
    const float* aR = xin    + (size_t)rowA * F + 2 * half;

    // Preload all A fragments up front: 32 pipelined global_load_b64.
    v2f fa[16], fb[16];
#pragma unroll
    for (int kk = 0; kk < 16; ++kk) {
        fa[kk] = *(const v2f*)(aL + kk * 4);
        fb[kk] = *(const v2f*)(aR + kk * 4);
    }
#pragma unroll
    for (int kk = 0; kk < 16; ++kk) { fa[kk].x *= sc; fa[kk].y *= sc; }

    v8f accL[4] = {};
    v8f accR[4] = {};

#pragma unroll
    for (int kk = 0; kk < 16; ++kk) {
        const v2f* bLp = &sWL2[(kk * 2 + half) * F + lr];
        const v2f* bRp = &sWR2[(kk * 2 + half) * F + lr];
#pragma unroll
        for (int t = 0; t < 4; ++t) {
            accL[t] = wmma_f32(fa[kk], bLp[t * 16], accL[t]);
            accR[t] = wmma_f32(fb[kk], bRp[t * 16], accR[t]);
        }
    }

    if (row0 + 16 <= N_NODES) {
        // fast path: whole tile in bounds, no per-element exec masking
#pragma unroll
        for (int t = 0; t < 4; ++t) {
            const float bv = bias[t * 16 + lr];
#pragma unroll
            for (int i = 0; i < 8; ++i) {
                const int r = row0 + i + 8 * half;   // C/D layout: M = vgpr + 8*half
                float v = accL[t][i] + accR[t][i] + bv;
                out[(size_t)r * F + t * 16 + lr] = fmaxf(v, 0.0f);
            }
        }
    } else {
#pragma unroll
        for (int t = 0; t < 4; ++t) {
            const float bv = bias[t * 16 + lr];
#pragma unroll
            for (int i = 0; i < 8; ++i) {
                const int r = row0 + i + 8 * half;
                if (r < N_NODES) {
                    float v = accL[t][i] + accR[t][i] + bv;
                    out[(size_t)r * F + t * 16 + lr] = fmaxf(v, 0.0f);
                }
            }
        }
    }
}

// ---------------- plain linear layer GEMM ----------------
// out = (relu?) (xin @ w + bias)
__global__ void __launch_bounds__(256)
lin_gemm_kernel(const float* __restrict__ xin, const float* __restrict__ w,
                const float* __restrict__ bias, float* __restrict__ out, int relu) {
    __shared__ v2f sW2[32 * F];
    stage_w_pairs(sW2, w);
    __syncthreads();

    const int lane = threadIdx.x & 31;
    const int wave = threadIdx.x >> 5;
    const int half = lane >> 4;
    const int lr   = lane & 15;
    const int row0 = blockIdx.x * 128 + wave * 16;

    int rowA = row0 + lr;
    if (rowA > N_NODES - 1) rowA = N_NODES - 1;
    const float* aP = xin + (size_t)rowA * F + 2 * half;

    v2f fa[16];
#pragma unroll
    for (int kk = 0; kk < 16; ++kk)
        fa[kk] = *(const v2f*)(aP + kk * 4);

    v8f acc[4] = {};
#pragma unroll
    for (int kk = 0; kk < 16; ++kk) {
        const v2f* bp = &sW2[(kk * 2 + half) * F + lr];
#pragma unroll
        for (int t = 0; t < 4; ++t)
            acc[t] = wmma_f32(fa[kk], bp[t * 16], acc[t]);
    }

    if (row0 + 16 <= N_NODES) {
#pragma unroll
        for (int t = 0; t < 4; ++t) {
            const float bv = bias[t * 16 + lr];
#pragma unroll
            for (int i = 0; i < 8; ++i) {
                const int r = row0 + i + 8 * half;
                float v = acc[t][i] + bv;
                if (relu) v = fmaxf(v, 0.0f);
                out[(size_t)r * F + t * 16 + lr] = v;
            }
        }
    } else {
#pragma unroll
        for (int t = 0; t < 4; ++t) {
            const float bv = bias[t * 16 + lr];
#pragma unroll
            for (int i = 0; i < 8; ++i) {
                const int r = row0 + i + 8 * half;
                if (r < N_NODES) {
                    float v = acc[t][i] + bv;
                    if (relu) v = fmaxf(v, 0.0f);
                    out[(size_t)r * F + t * 16 + lr] = v;
                }
            }
        }
    }
}

// ---------------- launch ----------------

extern "C" void kernel_launch(void* const* d_in, const int* in_sizes, int n_in,
                              void* d_out, int out_size, void* d_ws, size_t ws_size,
                              hipStream_t stream) {
    const float*     h   = (const float*)d_in[0];
    const long long* ei  = (const long long*)d_in[1];
    const long long* src = ei;
    const long long* dst = ei + NE;
    const float* w1l   = (const float*)d_in[2];
    const float* b1l   = (const float*)d_in[3];
    const float* w1r   = (const float*)d_in[4];
    const float* wlin1 = (const float*)d_in[5];
    const float* blin1 = (const float*)d_in[6];
    const float* w2l   = (const float*)d_in[7];
    const float* b2l   = (const float*)d_in[8];
    const float* w2r   = (const float*)d_in[9];
    const float* wlin2 = (const float*)d_in[10];
    const float* blin2 = (const float*)d_in[11];
    float* out = (float*)d_out;

    float* ws     = (float*)d_ws;
    float* cnt    = ws;                                  // N
    float* inv    = ws + N_NODES;                        // N
    float* summed = ws + 2 * (size_t)N_NODES;            // N*F
    float* x1     = summed + (size_t)N_NODES * F;        // N*F (reused for x3)
    float* x2     = x1 + (size_t)N_NODES * F;            // N*F

    const int TB = 256;
    const long nNF      = (long)N_NODES * F;
    const int  gZeroN   = (int)((N_NODES + TB - 1) / TB);
    const int  gZeroNF  = (int)((nNF + TB - 1) / TB);
    const int  gCount   = (NE + TB - 1) / TB;
    const int  gScatter = (int)(((long)NE * 16 + TB - 1) / TB);
    const int  gGemm    = (N_NODES + 127) / 128;

    // --- degrees (reused by both convs) ---
    zero_kernel<<<gZeroN, TB, 0, stream>>>(cnt, N_NODES);
    count_kernel<<<gCount, TB, 0, stream>>>(dst, cnt);
    inv_kernel<<<gZeroN, TB, 0, stream>>>(cnt, inv);

    // --- conv1: x1 = relu(mean(h)@w1l + b1l + h@w1r) ---
    zero_kernel<<<gZeroNF, TB, 0, stream>>>(summed, nNF);
    scatter_kernel<<<gScatter, TB, 0, stream>>>(h, src, dst, summed);
    sage_gemm_kernel<<<gGemm, TB, 0, stream>>>(summed, inv, h, w1l, w1r, b1l, x1);

    // --- lin1: x2 = relu(x1@wlin1 + blin1) ---
    lin_gemm_kernel<<<gGemm, TB, 0, stream>>>(x1, wlin1, blin1, x2, 1);

    // --- conv2: x1 = relu(mean(x2)@w2l + b2l + x2@w2r) ---
    zero_kernel<<<gZeroNF, TB, 0, stream>>>(summed, nNF);
    scatter_kernel<<<gScatter, TB, 0, stream>>>(x2, src, dst, summed);
    sage_gemm_kernel<<<gGemm, TB, 0, stream>>>(summed, inv, x2, w2l, w2r, b2l, x1);

    // --- lin2 (no relu) straight into d_out ---
    lin_gemm_kernel<<<gGemm, TB, 0, stream>>>(x1, wlin2, blin2, out, 0);
}